// GaussianHeatmapLoss_21723944583260
// MI455X (gfx1250) — compile-verified
//
#include <hip/hip_runtime.h>
#include <hip/hip_bf16.h>

typedef float v2f __attribute__((ext_vector_type(2)));
typedef float v8f __attribute__((ext_vector_type(8)));

#define HDIM 256
#define WDIM 256
#define NKP  448                    // B*K = 32*14
#define INV2S2 (1.0f / 4.5f)        // 1/(2*std^2), std = 1.5
#define EPSV 1e-6f
#define WAVES_PER_BLOCK 8
#define TILES_PER_WAVE 2            // (WDIM/16) / WAVES_PER_BLOCK

// One block = one keypoint's 16-row strip. 8 waves x 2 column tiles each.
// Each wave builds a 16x16 heatmap tile as a rank-1 outer product er (x) ec
// via V_WMMA_F32_16X16X4_F32 (only K=0 slice nonzero), then streams
// source in / heatmap out and accumulates the squared-error partial sum.
__global__ __launch_bounds__(256) void GaussianHeatmap_wmma_kernel(
    const float* __restrict__ src, const float* __restrict__ tgt,
    float* __restrict__ hm, float* __restrict__ partials)
{
    const int tid  = threadIdx.x;
    const int lane = tid & 31;              // wave32
    const int wave = tid >> 5;
    const int blk  = blockIdx.x;
    const int n    = blk >> 4;              // keypoint index
    const int row0 = (blk & 15) << 4;       // strip base row

    const float cx = tgt[2 * n + 0];        // compared against columns
    const float cy = tgt[2 * n + 1];        // compared against rows
    const bool valid = !((cx != cx) || (cy != cy));

    // Analytic grid max of the separable gaussian (nearest clamped grid point)
    const float hstar = fminf(fmaxf(rintf(cy), 0.0f), (float)(HDIM - 1));
    const float wstar = fminf(fmaxf(rintf(cx), 0.0f), (float)(WDIM - 1));
    const float dym = hstar - cy, dxm = wstar - cx;
    const float mx = __expf(-dym * dym * INV2S2) * __expf(-dxm * dxm * INV2S2);
    const float scale = (mx > EPSV) ? (1.0f / mx) : 1.0f;

    // A operand: K=0 column = er[row]*scale in VGPR0, lanes 0-15 (M = lane)
    const float dy = (float)(row0 + (lane & 15)) - cy;
    const float av = (valid && lane < 16) ? __expf(-dy * dy * INV2S2) * scale : 0.0f;
    v2f a; a.x = av; a.y = 0.0f;

    float lsum = 0.0f;
    const size_t base = (size_t)n * (HDIM * WDIM);

    for (int t = 0; t < TILES_PER_WAVE; ++t) {
        const int col0 = ((wave * TILES_PER_WAVE) + t) << 4;

        // B operand: K=0 row = ec[col] in VGPR0, lanes 0-15 (N = lane)
        const float dx = (float)(col0 + (lane & 15)) - cx;
        const float bv = (valid && lane < 16) ? __expf(-dx * dx * INV2S2) : 0.0f;
        v2f b; b.x = bv; b.y = 0.0f;

        v8f c = {0.f, 0.f, 0.f, 0.f, 0.f, 0.f, 0.f, 0.f};
        // (neg_a, A, neg_b, B, c_mod, C, reuse_a, reuse_b)
        c = __builtin_amdgcn_wmma_f32_16x16x4_f32(
                false, a, false, b, (short)0, c, false, false);

        // C/D layout: VGPR i -> row (i | i+8 by lane half), col = lane&15
        const int col   = col0 + (lane & 15);
        const int rbase = row0 + ((lane < 16) ? 0 : 8);
        #pragma unroll
        for (int i = 0; i < 8; ++i) {
            const size_t idx = base + (size_t)(rbase + i) * WDIM + col;
            const float h = c[i];
            const float d = src[idx] - h;
            lsum += d * d;
            hm[idx] = h;
        }
    }

    __shared__ float red[256];
    red[tid] = lsum;
    __syncthreads();
    for (int s = 128; s > 0; s >>= 1) {
        if (tid < s) red[tid] += red[tid + s];
        __syncthreads();
    }
    if (tid == 0) partials[blk] = red[0];
}

__global__ __launch_bounds__(1024) void GaussianHeatmap_reduce_kernel(
    const float* __restrict__ partials, int n, float inv_total,
    float* __restrict__ out)
{
    __shared__ float red[1024];
    float s = 0.0f;
    for (int i = threadIdx.x; i < n; i += 1024) s += partials[i];
    red[threadIdx.x] = s;
    __syncthreads();
    for (int k = 512; k > 0; k >>= 1) {
        if (threadIdx.x < k) red[threadIdx.x] += red[threadIdx.x + k];
        __syncthreads();
    }
    if (threadIdx.x == 0) out[0] = red[0] * inv_total;
}

extern "C" void kernel_launch(void* const* d_in, const int* in_sizes, int n_in,
                              void* d_out, int out_size, void* d_ws, size_t ws_size,
                              hipStream_t stream) {
    const float* src = (const float*)d_in[0];   // (32,14,256,256) f32
    const float* tgt = (const float*)d_in[1];   // (32,14,2) f32
    float* out = (float*)d_out;                 // [0] = loss, [1..] = heatmap
    float* hm  = out + 1;
    float* partials = (float*)d_ws;             // 7168 floats of scratch

    const int nblocks = NKP * (HDIM / 16);      // 7168
    GaussianHeatmap_wmma_kernel<<<nblocks, 256, 0, stream>>>(src, tgt, hm, partials);

    const float inv_total = 1.0f / (float)((size_t)NKP * HDIM * WDIM); // 1/29360128
    GaussianHeatmap_reduce_kernel<<<1, 1024, 0, stream>>>(partials, nblocks, inv_total, out);
}